// BioSignalEmbed_69011534512415
// MI455X (gfx1250) — compile-verified
//
#include <hip/hip_runtime.h>
#include <math.h>

typedef __attribute__((ext_vector_type(2))) float v2f;
typedef __attribute__((ext_vector_type(8))) float v8f;
typedef __attribute__((ext_vector_type(4))) int   v4i;

#define B_    8
#define T_    65536
#define C_    64
#define WIN_  64
#define HOP_  32
#define TW_   2047
#define ROWS_ (B_ * TW_)   /* 16376 */
#define HID_  512
#define KDIM_ 512

#if __has_builtin(__builtin_amdgcn_global_load_async_to_lds_b128) && \
    __has_builtin(__builtin_amdgcn_s_wait_asynccnt)
#define USE_ASYNC_LDS 1
typedef __attribute__((address_space(1))) v4i* gptr_b128;   // global (AS1) int4*
typedef __attribute__((address_space(3))) v4i* lptr_b128;   // LDS    (AS3) int4*
#endif

// ---------------------------------------------------------------------------
// Kernel 1: windowed spectral features + per-channel 7->8 projection
// One block per (b, t) window; the 64(time) x 64(chan) tile is contiguous.
// Tile staged via async global->LDS (no VGPR round-trip) when available.
// ---------------------------------------------------------------------------
__global__ __launch_bounds__(256) void feat_kernel(
    const float* __restrict__ signal, const float* __restrict__ chan_w,
    const float* __restrict__ chan_b, float* __restrict__ flat)
{
    __shared__ float tile[WIN_ * C_];   // [time][chan], 16 KB
    __shared__ float ctab[64];          // cos(2*pi*i/64)
    __shared__ float mag[C_][24];       // |X_k| for k=1..24

    const int tid = threadIdx.x;
    const int b   = blockIdx.x / TW_;
    const int t   = blockIdx.x - b * TW_;

    const float4* src4 = (const float4*)(signal + ((size_t)b * T_ + (size_t)t * HOP_) * C_);

#ifdef USE_ASYNC_LDS
    // 4096 floats = 1024 x b128 async copies straight into LDS (ASYNCcnt path)
#pragma unroll
    for (int p = 0; p < 4; ++p)
        __builtin_amdgcn_global_load_async_to_lds_b128(
            (gptr_b128)(const v4i*)(src4 + p * 256 + tid),
            (lptr_b128)(v4i*)((float4*)tile + p * 256 + tid),
            0, 0);
    if (tid < 64)
        ctab[tid] = __cosf((float)tid * (6.283185307179586f / 64.0f));
    __builtin_amdgcn_s_wait_asynccnt(0);
#else
    float4* tile4 = (float4*)tile;
#pragma unroll
    for (int p = 0; p < 4; ++p)
        tile4[p * 256 + tid] = src4[p * 256 + tid];
    if (tid < 64)
        ctab[tid] = __cosf((float)tid * (6.283185307179586f / 64.0f));
#endif
    __syncthreads();

    const int c = tid & 63;   // channel
    const int j = tid >> 6;   // bin group 0..3 (uniform per wave)

    float re[6] = {0.f,0.f,0.f,0.f,0.f,0.f};
    float im[6] = {0.f,0.f,0.f,0.f,0.f,0.f};
    float sum = 0.f, sumsq = 0.f;
    int kk[6], idx[6];
#pragma unroll
    for (int u = 0; u < 6; ++u) { kk[u] = 1 + j * 6 + u; idx[u] = 0; }

    for (int n = 0; n < WIN_; ++n) {
        float x = tile[n * C_ + c];                 // conflict-free: lanes hit consecutive banks
#pragma unroll
        for (int u = 0; u < 6; ++u) {
            float cs = ctab[idx[u]];                // uniform index per wave -> broadcast
            float sn = ctab[(idx[u] + 48) & 63];    // sin(th) = cos(th - pi/2)
            re[u] = fmaf(x,  cs, re[u]);
            im[u] = fmaf(-x, sn, im[u]);
            idx[u] = (idx[u] + kk[u]) & 63;
        }
        if (j == 0) { sum += x; sumsq = fmaf(x, x, sumsq); }
    }
#pragma unroll
    for (int u = 0; u < 6; ++u)
        mag[c][j * 6 + u] = sqrtf(re[u] * re[u] + im[u] * im[u]);
    __syncthreads();

    if (j == 0) {
        // band (0.5,4): empty -> 0 ; (4,8): k=1 ; (8,13): k=2,3 ;
        // (13,30): k=4..7 ; (30,100): k=8..24  (mag[] indexed by k-1)
        float f1 = mag[c][0];
        float f2 = 0.5f  * (mag[c][1] + mag[c][2]);
        float f3 = 0.25f * (mag[c][3] + mag[c][4] + mag[c][5] + mag[c][6]);
        float s4 = 0.f;
#pragma unroll
        for (int q = 7; q < 24; ++q) s4 += mag[c][q];
        float f4   = s4 * (1.0f / 17.0f);
        float mean = sum * (1.0f / 64.0f);
        float var  = (sumsq - sum * sum * (1.0f / 64.0f)) * (1.0f / 63.0f);
        float stdv = sqrtf(fmaxf(var, 0.f));
        float feats[7] = {0.f, f1, f2, f3, f4, mean, stdv};

        float* dst = flat + (size_t)blockIdx.x * (C_ * 8) + c * 8;
#pragma unroll
        for (int p = 0; p < 8; ++p) {
            float e = chan_b[c * 8 + p];
#pragma unroll
            for (int f = 0; f < 7; ++f)
                e = fmaf(feats[f], chan_w[(c * 7 + f) * 8 + p], e);
            dst[p] = e;
        }
    }
}

// ---------------------------------------------------------------------------
// Kernel 2: z = flat @ mix_w^T via V_WMMA_F32_16X16X4_F32, fused positional
// encoding and scatter into out[b, 1+t, :].
// Block = 32 rows x 512 cols; 8 waves, each wave 32M x 64N (2x4 WMMA tiles).
// ---------------------------------------------------------------------------
__global__ __launch_bounds__(256) void gemm_pe_kernel(
    const float* __restrict__ flat, const float* __restrict__ mix_w,
    float* __restrict__ out)
{
    const int lane = threadIdx.x & 31;
    const int wave = threadIdx.x >> 5;     // 0..7
    const int m0   = blockIdx.x * 32;
    const int n0   = wave * 64;

    const int lr = lane & 15;              // fragment row/col
    const int kb = (lane >> 4) * 2;        // K sub-offset {0, 2}

    v8f acc[2][4];
    const v8f vzero = {0.f,0.f,0.f,0.f,0.f,0.f,0.f,0.f};
#pragma unroll
    for (int s = 0; s < 2; ++s)
#pragma unroll
        for (int q = 0; q < 4; ++q) acc[s][q] = vzero;

    int ar[2];
#pragma unroll
    for (int s = 0; s < 2; ++s) {
        int m = m0 + s * 16 + lr;
        ar[s] = (m < ROWS_) ? m : (ROWS_ - 1);   // clamp: rows independent, store guarded
    }

    for (int k = 0; k < KDIM_; k += 4) {
        v2f a[2], bb[4];
#pragma unroll
        for (int s = 0; s < 2; ++s)
            a[s] = *(const v2f*)(flat + (size_t)ar[s] * KDIM_ + k + kb);
#pragma unroll
        for (int q = 0; q < 4; ++q) {
            int n = n0 + q * 16 + lr;
            bb[q] = *(const v2f*)(mix_w + (size_t)n * KDIM_ + k + kb);   // B[k][n] = mix_w[n][k]
        }
#pragma unroll
        for (int s = 0; s < 2; ++s)
#pragma unroll
            for (int q = 0; q < 4; ++q)
                acc[s][q] = __builtin_amdgcn_wmma_f32_16x16x4_f32(
                    false, a[s], false, bb[q], (short)0, acc[s][q], false, false);
    }

    // store with fused sinusoidal positional encoding
    const float negln = -9.210340371976184f / 256.0f;   // -ln(10000)/half
    const int laneHi  = (lane >> 4) * 8;
#pragma unroll
    for (int s = 0; s < 2; ++s) {
#pragma unroll
        for (int r = 0; r < 8; ++r) {
            int m = m0 + s * 16 + r + laneHi;
            if (m >= ROWS_) continue;
            int bi = m / TW_;
            int t  = m - bi * TW_;
#pragma unroll
            for (int q = 0; q < 4; ++q) {
                int n   = n0 + q * 16 + lr;
                int i   = n >> 1;
                float ang = (float)t * __expf((float)i * negln);
                float pe  = (n & 1) ? __cosf(ang) : __sinf(ang);
                size_t o  = ((size_t)bi * 2048 + 1 + (size_t)t) * HID_ + n;
                out[o] = acc[s][q][r] + pe;
            }
        }
    }
}

// ---------------------------------------------------------------------------
// Kernel 3: marker row (out[b, 0, :] = marker)
// ---------------------------------------------------------------------------
__global__ void marker_kernel(const float* __restrict__ marker, float* __restrict__ out)
{
    int i = blockIdx.x * blockDim.x + threadIdx.x;   // 0..4095
    int b = i >> 9, h = i & 511;
    out[(size_t)b * 2048 * HID_ + h] = marker[h];
}

extern "C" void kernel_launch(void* const* d_in, const int* in_sizes, int n_in,
                              void* d_out, int out_size, void* d_ws, size_t ws_size,
                              hipStream_t stream) {
    const float* signal = (const float*)d_in[0];
    const float* chan_w = (const float*)d_in[1];
    const float* chan_b = (const float*)d_in[2];
    const float* mix_w  = (const float*)d_in[3];
    const float* marker = (const float*)d_in[4];
    float* out  = (float*)d_out;
    float* flat = (float*)d_ws;   // ROWS_ * 512 floats = 33.5 MB scratch

    feat_kernel<<<ROWS_, 256, 0, stream>>>(signal, chan_w, chan_b, flat);
    gemm_pe_kernel<<<(ROWS_ + 31) / 32, 256, 0, stream>>>(flat, mix_w, out);
    marker_kernel<<<16, 256, 0, stream>>>(marker, out);
}